// GAT_214748365434
// MI455X (gfx1250) — compile-verified
//
#include <hip/hip_runtime.h>
#include <math.h>

// Problem constants (match reference)
#define NN      4096
#define NFEAT   512
#define NHID    64
#define NCLASS  128
#define NHEADS  8
#define ALPHA   0.2f
#define NEGV    (-9000000000000000.0f)

typedef __attribute__((ext_vector_type(2))) float v2f;
typedef __attribute__((ext_vector_type(8))) float v8f;

// ---------------------------------------------------------------------------
// Kernel: pack adjacency int32 (0/1) into a bitmask, 32 entries per word.
// ---------------------------------------------------------------------------
__global__ void pack_adj_kernel(const int* __restrict__ adj,
                                unsigned* __restrict__ bits, long nwords) {
    long w = (long)blockIdx.x * blockDim.x + threadIdx.x;
    if (w >= nwords) return;
    const int* p = adj + w * 32;
    unsigned m = 0u;
    #pragma unroll
    for (int b = 0; b < 32; ++b) m |= (p[b] > 0 ? 1u : 0u) << b;
    bits[w] = m;
}

// ---------------------------------------------------------------------------
// Kernel: fp32 WMMA GEMM  C[h] = A @ B[h]  (row-major), K/NB compile-time.
//   grid.x = M/16, grid.y = NB/64, grid.z = heads, block = 128 (4 waves).
// ---------------------------------------------------------------------------
template <int K, int NB>
__global__ __launch_bounds__(128) void wmma_gemm_f32(
        const float* __restrict__ A,
        const float* __restrict__ B, long bStride,
        float* __restrict__ C, long cStride) {
    const int wave = threadIdx.x >> 5;
    const int lane = threadIdx.x & 31;
    const int hi   = lane >> 4;
    const int lo   = lane & 15;
    const int row0 = blockIdx.x * 16;
    const int col0 = (blockIdx.y * 4 + wave) * 16;
    const int head = blockIdx.z;

    const float* __restrict__ arow  = A + (long)(row0 + lo) * K + 2 * hi;
    const float* __restrict__ bbase = B + (long)head * bStride + 2 * hi * NB + col0 + lo;

    v8f acc = {};
    for (int k = 0; k < K; k += 4) {
        v2f a;
        a.x = arow[k];
        a.y = arow[k + 1];
        v2f b;
        b.x = bbase[k * NB];
        b.y = bbase[k * NB + NB];
        acc = __builtin_amdgcn_wmma_f32_16x16x4_f32(
            false, a, false, b, (short)0, acc, false, false);
    }

    float* cp = C + (long)head * cStride + (long)(row0 + 8 * hi) * NB + col0 + lo;
    #pragma unroll
    for (int v = 0; v < 8; ++v) cp[v * NB] = acc[v];
}

// ---------------------------------------------------------------------------
// Kernel: fi[h][n] = H[h][n][:] . a1[h],  fj[h][n] = H[h][n][:] . a2[h]
// ---------------------------------------------------------------------------
__global__ void fused_fifj(const float* __restrict__ H, long hStride, int F,
                           const float* __restrict__ a, int aStride,
                           float* __restrict__ fi, float* __restrict__ fj, int n) {
    const int head = blockIdx.y;
    const int nd   = blockIdx.x * blockDim.x + threadIdx.x;
    if (nd >= n) return;
    const float* hp = H + (long)head * hStride + (long)nd * F;
    const float* a1 = a + (long)head * aStride;
    const float* a2 = a1 + F;
    float s1 = 0.f, s2 = 0.f;
    for (int o = 0; o < F; ++o) {
        float hv = hp[o];
        s1 += hv * a1[o];
        s2 += hv * a2[o];
    }
    fi[(long)head * n + nd] = s1;
    fj[(long)head * n + nd] = s2;
}

// ---------------------------------------------------------------------------
// Kernel: split-j flash-style GAT attention + aggregation, R row-tiles/block.
//   grid.x = n/(16R) row-tile groups, grid.y = heads, block = JW waves.
//   Each wave owns a contiguous j-chunk with R independent (m,l,acc[16xF])
//   online-softmax states; B fragments are loaded once per j-tile and reused
//   (in registers) by all R row-tiles' V_WMMA_F32_16X16X4_F32 chains, halving
//   (R=2) the dominant L2 traffic. Per-wave partials are merged through LDS
//   in R fixed-order passes (deterministic, no float atomics).
// ---------------------------------------------------------------------------
template <int F, int JW, int R, bool ELU>
__global__ __launch_bounds__(JW * 32) void gat_attn(
        const float* __restrict__ Hfeat, long hStride,
        const float* __restrict__ fi, const float* __restrict__ fj,
        const unsigned* __restrict__ bits,
        float* __restrict__ Out, int ldo, int headColMul, int n) {
    constexpr int NC = F / 16;            // 16-col output chunks
    const int wave = threadIdx.x >> 5;
    const int lane = threadIdx.x & 31;
    const int hi   = lane >> 4;
    const int lo   = lane & 15;
    const int head = blockIdx.y;
    const int i0   = blockIdx.x * (16 * R);
    const int nwords = n >> 5;

    const float* __restrict__ Hh  = Hfeat + (long)head * hStride;
    const float* __restrict__ fih = fi + (long)head * n;
    const float* __restrict__ fjh = fj + (long)head * n;
    const float* __restrict__ bbase = Hh + 2 * hi * F + lo;

    float fiv[R];
    const unsigned* __restrict__ brow[R];
    #pragma unroll
    for (int r = 0; r < R; ++r) {
        fiv[r]  = fih[i0 + 16 * r + lo];            // lane tracks row `lo`
        brow[r] = bits + (long)(i0 + 16 * r + lo) * nwords;
    }

    const int chunk = n / JW;
    const int jbeg  = wave * chunk;
    const int jend  = jbeg + chunk;

    float m[R], l[R];
    v8f acc[R][NC];
    #pragma unroll
    for (int r = 0; r < R; ++r) {
        m[r] = -INFINITY;
        l[r] = 0.f;
        #pragma unroll
        for (int c = 0; c < NC; ++c) acc[r][c] = (v8f){};
    }

    for (int j0 = jbeg; j0 < jend; j0 += 32) {
        unsigned wmask[R];
        #pragma unroll
        for (int r = 0; r < R; ++r) wmask[r] = brow[r][j0 >> 5];

        #pragma unroll
        for (int half = 0; half < 2; ++half) {
            const int jb = j0 + 16 * half;
            const int bitbase = 16 * half;
            const float* __restrict__ fjp = fjh + jb + 2 * hi;

            // fj tile values (shared across all R row-tiles)
            float fjv[8];
            #pragma unroll
            for (int s = 0; s < 4; ++s)
                #pragma unroll
                for (int t = 0; t < 2; ++t) fjv[2 * s + t] = fjp[4 * s + t];

            // per row-tile: masked leaky scores -> online softmax -> p8
            float p8[R][8];
            #pragma unroll
            for (int r = 0; r < R; ++r) {
                float e8[8];
                float tmax = -INFINITY;
                #pragma unroll
                for (int s = 0; s < 4; ++s) {
                    #pragma unroll
                    for (int t = 0; t < 2; ++t) {
                        const int c = 4 * s + 2 * hi + t;
                        float v = fiv[r] + fjv[2 * s + t];
                        v = v > 0.f ? v : ALPHA * v;           // leaky relu
                        const float e = ((wmask[r] >> (bitbase + c)) & 1u) ? v : NEGV;
                        e8[2 * s + t] = e;
                        tmax = fmaxf(tmax, e);
                    }
                }
                tmax = fmaxf(tmax, __shfl_xor(tmax, 16, 32));  // join half-rows
                const float mnew = fmaxf(m[r], tmax);

                float psum = 0.f;
                #pragma unroll
                for (int q = 0; q < 8; ++q) {
                    p8[r][q] = __expf(e8[q] - mnew);
                    psum += p8[r][q];
                }
                psum += __shfl_xor(psum, 16, 32);

                if (__any(mnew > m[r])) {                      // rare rescale
                    const float scale = __expf(m[r] - mnew);
                    float sv[8];
                    #pragma unroll
                    for (int v = 0; v < 8; ++v) sv[v] = __shfl(scale, v + 8 * hi, 32);
                    #pragma unroll
                    for (int c = 0; c < NC; ++c)
                        #pragma unroll
                        for (int v = 0; v < 8; ++v) acc[r][c][v] *= sv[v];
                    l[r] = l[r] * scale + psum;
                    m[r] = mnew;
                } else {
                    l[r] += psum;
                }
            }

            // B fragments loaded once, reused by all R row-tiles' WMMAs
            const float* __restrict__ bp = bbase + jb * F;
            #pragma unroll
            for (int s = 0; s < 4; ++s) {
                const float* __restrict__ bs = bp + 4 * s * F;
                #pragma unroll
                for (int c = 0; c < NC; ++c) {
                    v2f b;
                    b.x = bs[16 * c];
                    b.y = bs[16 * c + F];
                    #pragma unroll
                    for (int r = 0; r < R; ++r) {
                        v2f a;
                        a.x = p8[r][2 * s];
                        a.y = p8[r][2 * s + 1];
                        acc[r][c] = __builtin_amdgcn_wmma_f32_16x16x4_f32(
                            false, a, false, b, (short)0, acc[r][c], false, false);
                    }
                }
            }
        }
    }

    // ---- deterministic cross-wave softmax merge (R passes over one buffer) --
    __shared__ float m_lds[R][JW * 16];
    __shared__ float l_lds[R][JW * 16];
    __shared__ float L_lds[16];
    __shared__ float accbuf[JW * 16 * F];

    #pragma unroll
    for (int r = 0; r < R; ++r) {
        if (hi == 0) {                    // lanes 0..15 carry row `lo` state
            m_lds[r][wave * 16 + lo] = m[r];
            l_lds[r][wave * 16 + lo] = l[r];
        }
    }
    __syncthreads();

    constexpr int TOT = 16 * F;
    constexpr int NT  = JW * 32;
    #pragma unroll
    for (int r = 0; r < R; ++r) {
        float Mrow = -INFINITY;
        #pragma unroll
        for (int w = 0; w < JW; ++w) Mrow = fmaxf(Mrow, m_lds[r][w * 16 + lo]);
        float Lrow = 0.f;
        #pragma unroll
        for (int w = 0; w < JW; ++w)
            Lrow += l_lds[r][w * 16 + lo] * __expf(m_lds[r][w * 16 + lo] - Mrow);
        if (wave == 0 && hi == 0) L_lds[lo] = Lrow;

        const float fown = __expf(m[r] - Mrow);   // merge factor, row `lo`
        float fv[8];
        #pragma unroll
        for (int v = 0; v < 8; ++v) fv[v] = __shfl(fown, v + 8 * hi, 32);

        float* ab = accbuf + (wave * 16 + 8 * hi) * F + lo;
        #pragma unroll
        for (int c = 0; c < NC; ++c)
            #pragma unroll
            for (int v = 0; v < 8; ++v)
                ab[v * F + 16 * c] = acc[r][c][v] * fv[v];
        __syncthreads();

        // fixed-order reduction over the JW partials + store rows of tile r
        float* outBase = Out + (long)(i0 + 16 * r) * ldo + head * headColMul;
        #pragma unroll
        for (int e = 0; e < TOT / NT; ++e) {
            const int flat = e * NT + threadIdx.x;
            const int row = flat / F;
            const int col = flat % F;
            float val = 0.f;
            #pragma unroll
            for (int w = 0; w < JW; ++w) val += accbuf[(w * 16 + row) * F + col];
            float rr = val / L_lds[row];
            if (ELU) rr = rr > 0.f ? rr : (__expf(rr) - 1.0f);
            outBase[(long)row * ldo + col] = rr;
        }
        if (r + 1 < R) __syncthreads();   // protect accbuf/L_lds reuse
    }
}

// ---------------------------------------------------------------------------
// Kernel: row-wise log_softmax over 128 columns; one wave per row.
// ---------------------------------------------------------------------------
__global__ void logsoftmax128(const float* __restrict__ in,
                              float* __restrict__ out, int n) {
    const int wave = threadIdx.x >> 5;
    const int lane = threadIdx.x & 31;
    const int row  = blockIdx.x * (blockDim.x >> 5) + wave;
    if (row >= n) return;
    const float* rp = in + (long)row * NCLASS;
    float v[4];
    float mx = -INFINITY;
    #pragma unroll
    for (int q = 0; q < 4; ++q) {
        v[q] = rp[lane + 32 * q];
        mx = fmaxf(mx, v[q]);
    }
    #pragma unroll
    for (int off = 16; off >= 1; off >>= 1) mx = fmaxf(mx, __shfl_xor(mx, off, 32));
    float s = 0.f;
    #pragma unroll
    for (int q = 0; q < 4; ++q) s += __expf(v[q] - mx);
    #pragma unroll
    for (int off = 16; off >= 1; off >>= 1) s += __shfl_xor(s, off, 32);
    const float ls = __logf(s);
    #pragma unroll
    for (int q = 0; q < 4; ++q) out[(long)row * NCLASS + lane + 32 * q] = v[q] - mx - ls;
}

// ---------------------------------------------------------------------------
extern "C" void kernel_launch(void* const* d_in, const int* in_sizes, int n_in,
                              void* d_out, int out_size, void* d_ws, size_t ws_size,
                              hipStream_t stream) {
    const float* x     = (const float*)d_in[0];   // [N, NFEAT]
    const int*   adj   = (const int*)d_in[1];     // [N, N]
    const float* W     = (const float*)d_in[2];   // [NHEADS, NFEAT, NHID]
    const float* a     = (const float*)d_in[3];   // [NHEADS, 2*NHID]
    const float* W_out = (const float*)d_in[4];   // [NHEADS*NHID, NCLASS]
    const float* a_out = (const float*)d_in[5];   // [2*NCLASS]
    float* out = (float*)d_out;                   // [N, NCLASS]

    // ---- workspace carve-up ----
    char* ws = (char*)d_ws;
    unsigned* bits = (unsigned*)ws;                       // N*N/32 words (2MB)
    ws += (size_t)NN * (NN / 32) * sizeof(unsigned);
    float* h1 = (float*)ws;                               // [8,4096,64] (8MB)
    ws += (size_t)NHEADS * NN * NHID * sizeof(float);
    float* fi1 = (float*)ws; ws += (size_t)NHEADS * NN * sizeof(float);
    float* fj1 = (float*)ws; ws += (size_t)NHEADS * NN * sizeof(float);
    float* xcat = (float*)ws;                             // [4096,512] (8MB)
    ws += (size_t)NN * (NHEADS * NHID) * sizeof(float);
    float* h2 = (float*)ws;                               // [4096,128] (2MB)
    ws += (size_t)NN * NCLASS * sizeof(float);
    float* fi2 = (float*)ws; ws += (size_t)NN * sizeof(float);
    float* fj2 = (float*)ws; ws += (size_t)NN * sizeof(float);
    float* outpre = (float*)ws;                           // [4096,128] (2MB)

    // 1) pack adjacency into bitmask (adj read exactly once)
    {
        const long nwords = (long)NN * (NN / 32);
        pack_adj_kernel<<<dim3((nwords + 255) / 256), dim3(256), 0, stream>>>(adj, bits, nwords);
    }

    // 2) layer-1 projection: h1[h] = x @ W[h]   (M=4096,K=512,N=64, 8 heads)
    wmma_gemm_f32<NFEAT, NHID><<<dim3(NN / 16, 1, NHEADS), dim3(128), 0, stream>>>(
        x, W, (long)NFEAT * NHID, h1, (long)NN * NHID);

    // 3) fi/fj for layer 1
    fused_fifj<<<dim3(NN / 256, NHEADS), dim3(256), 0, stream>>>(
        h1, (long)NN * NHID, NHID, a, 2 * NHID, fi1, fj1, NN);

    // 4) layer-1 attention + aggregate -> xcat (concat-by-head layout)
    gat_attn<NHID, 4, 2, false><<<dim3(NN / 32, NHEADS), dim3(128), 0, stream>>>(
        h1, (long)NN * NHID, fi1, fj1, bits,
        xcat, NHEADS * NHID, NHID, NN);

    // 5) layer-2 projection: h2 = xcat @ W_out  (M=4096,K=512,N=128)
    wmma_gemm_f32<NHEADS * NHID, NCLASS><<<dim3(NN / 16, NCLASS / 64, 1), dim3(128), 0, stream>>>(
        xcat, W_out, 0, h2, 0);

    // 6) fi/fj for layer 2
    fused_fifj<<<dim3(NN / 256, 1), dim3(256), 0, stream>>>(
        h2, 0, NCLASS, a_out, 2 * NCLASS, fi2, fj2, NN);

    // 7) layer-2 attention + aggregate + ELU -> outpre
    gat_attn<NCLASS, 4, 2, true><<<dim3(NN / 32, 1), dim3(128), 0, stream>>>(
        h2, 0, fi2, fj2, bits,
        outpre, NCLASS, 0, NN);

    // 8) log_softmax over classes -> d_out
    logsoftmax128<<<dim3(NN / 4), dim3(128), 0, stream>>>(outpre, out, NN);
}